// LaGCF_84164179132782
// MI455X (gfx1250) — compile-verified
//
#include <hip/hip_runtime.h>

// ---------------------------------------------------------------------------
// LaGCF / LightGCN-style propagation for MI455X (gfx1250, wave32).
//
// Pipeline per call:
//   E = acc = concat(user_emb, item_emb)                  (init_emb)
//   for l in 0..2:
//     S = 0                                                (hipMemsetAsync)
//     S = A * E            (edge-parallel scatter SpMM)    (spmm_kernel)
//     h = S + emb0 ; E = th*(h@W_l) + (1-th)*h ; acc += E  (combine_kernel, WMMA f32)
//   out = gather(acc/4, users/pos/neg)                     (gather_kernel)
//
// Scratch layout in d_ws: [E | S | acc], each 300000*64 f32 (76.8 MB).
// ---------------------------------------------------------------------------

#define N_USERS 100000
#define N_ITEMS 200000
#define N_NODES 300000
#define EMB     64
#define NNZ     9600000
#define BATCH   8192
#define N_LAYERS 3

typedef __attribute__((ext_vector_type(2))) float v2f;
typedef __attribute__((ext_vector_type(8))) float v8f;

// ---------------------------------------------------------------------------
// E = acc = concat(user_emb, item_emb), vectorized float4 (16 float4 per node)
// ---------------------------------------------------------------------------
__global__ void init_emb_kernel(const float4* __restrict__ u,
                                const float4* __restrict__ it,
                                float4* __restrict__ E,
                                float4* __restrict__ acc) {
    long i = (long)blockIdx.x * blockDim.x + threadIdx.x;
    const long total = (long)N_NODES * (EMB / 4);
    if (i >= total) return;
    long node = i >> 4;                                   // 16 float4 per node
    float4 v = (node < N_USERS) ? u[i] : it[i - (long)N_USERS * (EMB / 4)];
    E[i] = v;
    acc[i] = v;
}

// ---------------------------------------------------------------------------
// SpMM: one wave32 per edge. The 3 index loads are wave-uniform; lanes cover
// the 64-wide embedding row as float2 (coalesced 256B gather from L2-resident
// E) and scatter with native f32 atomics into L2-resident S.
// ---------------------------------------------------------------------------
__global__ void __launch_bounds__(256)
spmm_kernel(const int* __restrict__ rows, const int* __restrict__ cols,
            const float* __restrict__ vals, const float* __restrict__ E,
            float* __restrict__ S) {
    long e = (long)blockIdx.x * 8 + (threadIdx.x >> 5);   // 8 waves / block
    if (e >= NNZ) return;
    int lane = threadIdx.x & 31;
    int r = rows[e];
    int c = cols[e];
    float v = vals[e];
    const float2 x = *(const float2*)(E + (long)c * EMB + lane * 2);
    float* dst = S + (long)r * EMB + lane * 2;
    unsafeAtomicAdd(dst + 0, v * x.x);                    // global_atomic_add_f32
    unsafeAtomicAdd(dst + 1, v * x.y);
}

// ---------------------------------------------------------------------------
// Combine: per 16-node tile, h = S + emb0; G = h @ W (full-f32 WMMA,
// V_WMMA_F32_16X16X4_F32, 16 k-steps); E = th*G + (1-th)*h; acc += E.
// 4 waves/block, wave j owns output columns [16j,16j+16).
//
// LDS strides: h stride 68 -> bank(m*68+k)%64 = (4m+k)%64, both half-waves
// (k vs k+2) hit disjoint bank sets (0 vs 2 mod 4). W stride 72 -> rows 2 apart
// differ by 16 banks, the two half-waves use disjoint 16-bank groups.
// ---------------------------------------------------------------------------
#define HSTR 68
#define WSTR 72

__global__ void __launch_bounds__(128)
combine_kernel(const float* __restrict__ S, const float* __restrict__ u,
               const float* __restrict__ it, const float* __restrict__ Wl,
               float* __restrict__ E, float* __restrict__ acc, float theta) {
    __shared__ float wlds[64 * WSTR];
    __shared__ float hlds[16 * HSTR];
    const int tid  = threadIdx.x;
    const int wave = tid >> 5;
    const int lane = tid & 31;
    const int half = lane >> 4;          // 0: lanes 0-15, 1: lanes 16-31
    const int l16  = lane & 15;

    // Stage W (64x64) into LDS once per block.
    for (int i = tid; i < 64 * 64; i += 128) {
        int k = i >> 6, n = i & 63;
        wlds[k * WSTR + n] = Wl[i];
    }

    const int nTiles = N_NODES / 16;     // 18750, exact
    for (int t = blockIdx.x; t < nTiles; t += gridDim.x) {
        const long base = (long)t * 16 * EMB;
        __syncthreads();                 // hlds reuse guard (+W ready, 1st iter)
        for (int i = tid; i < 16 * EMB; i += 128) {
            int rr = i >> 6, cc = i & 63;
            long node = (long)t * 16 + rr;
            float e0 = (node < N_USERS) ? u[node * EMB + cc]
                                        : it[(node - N_USERS) * EMB + cc];
            hlds[rr * HSTR + cc] = S[base + i] + e0;      // h = S + emb0
        }
        __syncthreads();

        v8f d = {};                      // 16x16 f32 accumulator tile
        #pragma unroll
        for (int k0 = 0; k0 < EMB; k0 += 4) {
            const int ka = k0 + 2 * half;                 // half-wave K offset
            v2f a, b;
            a.x = hlds[l16 * HSTR + ka];                  // A: M=l16, K=ka
            a.y = hlds[l16 * HSTR + ka + 1];              //    K=ka+1
            b.x = wlds[ka * WSTR + wave * 16 + l16];      // B: K=ka,   N=l16
            b.y = wlds[(ka + 1) * WSTR + wave * 16 + l16];
            d = __builtin_amdgcn_wmma_f32_16x16x4_f32(
                    false, a, false, b, (short)0, d, false, false);
        }

        // D row r lives at M = r + 8*half, N = wave*16 + l16.
        const int col = wave * 16 + l16;
        #pragma unroll
        for (int r = 0; r < 8; ++r) {
            int row = r + 8 * half;
            float h  = hlds[row * HSTR + col];
            float nv = theta * d[r] + (1.0f - theta) * h;
            long gi  = base + (long)row * EMB + col;
            E[gi]    = nv;
            acc[gi] += nv;
        }
    }
}

// ---------------------------------------------------------------------------
// Gather rows of acc/4 for users / pos_items / neg_items.
// ---------------------------------------------------------------------------
__global__ void gather_kernel(const float* __restrict__ acc,
                              const int* __restrict__ users,
                              const int* __restrict__ pos,
                              const int* __restrict__ neg,
                              float* __restrict__ out) {
    long i = (long)blockIdx.x * blockDim.x + threadIdx.x;
    const long per = (long)BATCH * EMB;
    if (i >= 3 * per) return;
    int seg = (int)(i / per);
    long j  = i - (long)seg * per;
    int b = (int)(j >> 6);
    int c = (int)(j & 63);
    long node;
    if      (seg == 0) node = users[b];
    else if (seg == 1) node = (long)N_USERS + pos[b];
    else               node = (long)N_USERS + neg[b];
    out[i] = acc[node * EMB + c] * 0.25f;
}

// ---------------------------------------------------------------------------
extern "C" void kernel_launch(void* const* d_in, const int* in_sizes, int n_in,
                              void* d_out, int out_size, void* d_ws, size_t ws_size,
                              hipStream_t stream) {
    const float* user_emb = (const float*)d_in[0];
    const float* item_emb = (const float*)d_in[1];
    const float* W        = (const float*)d_in[2];   // 3 x 64 x 64
    const float* vals     = (const float*)d_in[3];
    const int*   rows     = (const int*)d_in[4];
    const int*   cols     = (const int*)d_in[5];
    const int*   users    = (const int*)d_in[6];
    const int*   pos      = (const int*)d_in[7];
    const int*   neg      = (const int*)d_in[8];
    float* out = (float*)d_out;

    const size_t nodeFloats = (size_t)N_NODES * EMB;  // 19.2M floats each
    float* E   = (float*)d_ws;
    float* S   = E + nodeFloats;
    float* acc = S + nodeFloats;

    {   // E = acc = concat(user_emb, item_emb)
        long total = (long)N_NODES * (EMB / 4);
        init_emb_kernel<<<(int)((total + 255) / 256), 256, 0, stream>>>(
            (const float4*)user_emb, (const float4*)item_emb,
            (float4*)E, (float4*)acc);
    }

    // theta_l = log(1/(l+1) + 1)
    const float thetas[N_LAYERS] = {0.69314718055994531f,   // log 2
                                    0.40546510810816438f,   // log 1.5
                                    0.28768207245178085f};  // log 4/3
    for (int l = 0; l < N_LAYERS; ++l) {
        hipMemsetAsync(S, 0, nodeFloats * sizeof(float), stream);
        spmm_kernel<<<NNZ / 8, 256, 0, stream>>>(rows, cols, vals, E, S);
        combine_kernel<<<4096, 128, 0, stream>>>(
            S, user_emb, item_emb, W + (size_t)l * EMB * EMB, E, acc, thetas[l]);
    }

    gather_kernel<<<(3 * BATCH * EMB + 255) / 256, 256, 0, stream>>>(
        acc, users, pos, neg, out);
}